// WavetableSynth_68908455297580
// MI455X (gfx1250) — compile-verified
//
#include <hip/hip_runtime.h>

#define MODEL_DIM 128
#define N_FRAMES  128
#define N_SAMPLES 32768
#define N_TABLES  16
#define TABLE_SIZE 512

typedef __attribute__((ext_vector_type(2))) float v2f;
typedef __attribute__((ext_vector_type(4))) float v4f;
typedef __attribute__((ext_vector_type(8))) float v8f;

// ---------------------------------------------------------------------------
// Wave-cooperative 16xN = 16x128 @ 128xN GEMM using V_WMMA_F32_16X16X4_F32.
// A (16x128, row-major, stride 128) lives in LDS; W (128xN row-major) + bias in
// global. Each wave owns output tiles of 16 columns; K consumed in 32 steps of 4.
// ISA fragment layouts (CDNA5 ISA §7.12.2):
//   A 16x4 : lanes 0-15 -> M=lane, V0/V1 = K0/K1 ; lanes 16-31 -> K2/K3
//   B 4x16 : lanes 0-15 -> N=lane, V0/V1 = K0/K1 ; lanes 16-31 -> K2/K3
//   C/D    : VGPR r -> row r (lanes 0-15) / row r+8 (lanes 16-31), col = lane%16
// ---------------------------------------------------------------------------
__device__ __forceinline__ void gemm16(const float* sA, float* sC,
                                       const float* __restrict__ W,
                                       const float* __restrict__ bias,
                                       int N, bool lrelu, int tid)
{
  const int K = MODEL_DIM;
  const int wave = tid >> 5;
  const int lane = tid & 31;
  const int half = lane >> 4;   // 0: lanes 0-15, 1: lanes 16-31
  const int l    = lane & 15;
  const int ntiles = N >> 4;
  for (int tile = wave; tile < ntiles; tile += 16) {   // uniform per wave: EXEC stays all-1s
    v8f acc = {0.f,0.f,0.f,0.f,0.f,0.f,0.f,0.f};
    const int n = tile * 16 + l;
    for (int k0 = 0; k0 < K; k0 += 4) {
      v2f a, b;
      a[0] = sA[l * K + k0 + 2 * half + 0];
      a[1] = sA[l * K + k0 + 2 * half + 1];
      b[0] = W[(size_t)(k0 + 2 * half + 0) * N + n];
      b[1] = W[(size_t)(k0 + 2 * half + 1) * N + n];
      acc = __builtin_amdgcn_wmma_f32_16x16x4_f32(false, a, false, b,
                                                  (short)0, acc, false, false);
    }
    const float bv = bias[n];
#pragma unroll
    for (int r = 0; r < 8; ++r) {
      float v = acc[r] + bv;
      if (lrelu) v = v > 0.f ? v : 0.2f * v;
      sC[(r + 8 * half) * N + n] = v;
    }
  }
}

// ---------------------------------------------------------------------------
// Kernel 1: all three MLP heads (WMMA), table choice, table normalization,
// squared env/freq frames. Single block, 512 threads (16 waves).
// ---------------------------------------------------------------------------
__global__ __launch_bounds__(512)
void heads_kernel(const float* __restrict__ x, const float* __restrict__ wavetables,
                  const float* tc_w1, const float* tc_b1, const float* tc_w2, const float* tc_b2,
                  const float* tc_w3, const float* tc_b3,
                  const float* env_w1, const float* env_b1, const float* env_w2, const float* env_b2,
                  const float* env_w3, const float* env_b3,
                  const float* fr_w1, const float* fr_b1, const float* fr_w2, const float* fr_b2,
                  const float* fr_w3, const float* fr_b3,
                  float* __restrict__ ws_values, float* __restrict__ ws_envf,
                  float* __restrict__ ws_frqf, float* __restrict__ ws_table)
{
  __shared__ float sX[16 * MODEL_DIM];
  __shared__ float sH1[16 * MODEL_DIM];
  __shared__ float sH2[16 * MODEL_DIM];
  __shared__ float sO[16 * 256];          // big enough for the N=256 tc head
  __shared__ float sScore[8 * N_TABLES];
  __shared__ int   sIdx[8];

  const int tid = threadIdx.x;

  // stage x (8x128) zero-padded to 16 rows
  for (int i = tid; i < 16 * MODEL_DIM; i += 512) {
    const int row = i >> 7, col = i & 127;
    sX[i] = (row < 8) ? x[row * MODEL_DIM + col] : 0.f;
  }
  __syncthreads();

  // ---- table-choice MLP: 128 -> 128 -> 128 -> 256 ----
  gemm16(sX,  sH1, tc_w1, tc_b1, 128, true,  tid); __syncthreads();
  gemm16(sH1, sH2, tc_w2, tc_b2, 128, true,  tid); __syncthreads();
  gemm16(sH2, sO,  tc_w3, tc_b3, 256, false, tid); __syncthreads();

  // scores[b][t] = mean_j tc[b][t*16+j]
  if (tid < 8 * N_TABLES) {
    const int b = tid >> 4, t = tid & 15;
    float s = 0.f;
    for (int j = 0; j < 16; ++j) s += sO[b * 256 + t * 16 + j];
    sScore[tid] = s * (1.f / 16.f);
  }
  __syncthreads();
  if (tid < 8) {
    int best = 0; float bv = sScore[tid * N_TABLES];
    for (int t = 1; t < N_TABLES; ++t) {
      const float v = sScore[tid * N_TABLES + t];
      if (v > bv) { bv = v; best = t; }     // first-max tie-break (jnp.argmax)
    }
    sIdx[tid] = best;
    ws_values[tid] = bv;
  }
  __syncthreads();

  // normalized selected table per batch item (one wave per b)
  {
    const int wave = tid >> 5, lane = tid & 31;
    if (wave < 8) {
      const float* tab = wavetables + (size_t)sIdx[wave] * TABLE_SIZE;
      float mx = -3.402823466e38f;
      for (int j = lane; j < TABLE_SIZE; j += 32) mx = fmaxf(mx, tab[j]);
      for (int off = 16; off > 0; off >>= 1) mx = fmaxf(mx, __shfl_xor(mx, off, 32));
      const float inv = 1.f / (mx + 1e-8f);
      for (int j = lane; j < TABLE_SIZE; j += 32)
        ws_table[wave * TABLE_SIZE + j] = tab[j] * inv;
    }
  }
  __syncthreads();

  // ---- envelope MLP: 128 -> 128 -> 128 -> 128, frames squared ----
  gemm16(sX,  sH1, env_w1, env_b1, 128, true,  tid); __syncthreads();
  gemm16(sH1, sH2, env_w2, env_b2, 128, true,  tid); __syncthreads();
  gemm16(sH2, sO,  env_w3, env_b3, 128, false, tid); __syncthreads();
  if (tid < 8 * N_FRAMES) { const float v = sO[tid]; ws_envf[tid] = v * v; }
  __syncthreads();

  // ---- frequency MLP: 128 -> 128 -> 128 -> 128, frames squared ----
  gemm16(sX,  sH1, fr_w1, fr_b1, 128, true,  tid); __syncthreads();
  gemm16(sH1, sH2, fr_w2, fr_b2, 128, true,  tid); __syncthreads();
  gemm16(sH2, sO,  fr_w3, fr_b3, 128, false, tid); __syncthreads();
  if (tid < 8 * N_FRAMES) { const float v = sO[tid]; ws_frqf[tid] = v * v; }
}

// ---------------------------------------------------------------------------
// Kernel 2: linear interp (align_corners=False) of freq/env frames to 32768
// samples, parallel scan for phase = frac(cumsum(freq)). One block per batch.
// ---------------------------------------------------------------------------
__global__ __launch_bounds__(1024)
void scan_kernel(const float* __restrict__ ws_envf, const float* __restrict__ ws_frqf,
                 float* __restrict__ ws_phase, float* __restrict__ ws_envs)
{
  __shared__ float sEnvF[N_FRAMES];
  __shared__ float sFrqF[N_FRAMES];
  __shared__ float sScan[1024];

  const int b = blockIdx.x;
  const int t = threadIdx.x;
  if (t < N_FRAMES) {
    sEnvF[t] = ws_envf[b * N_FRAMES + t];
    sFrqF[t] = ws_frqf[b * N_FRAMES + t];
  }
  __syncthreads();

  const int   SPT   = N_SAMPLES / 1024;                   // 32 samples per thread
  const float scale = (float)N_FRAMES / (float)N_SAMPLES; // 1/256 exact
  const int   s0    = t * SPT;

  float sum = 0.f;
  for (int i = 0; i < SPT; ++i) {
    const int s = s0 + i;
    float pos = ((float)s + 0.5f) * scale - 0.5f;
    pos = fminf(fmaxf(pos, 0.f), (float)(N_FRAMES - 1));
    const int lo = (int)pos;
    const int hi = min(lo + 1, N_FRAMES - 1);
    const float w = pos - (float)lo;
    sum += sFrqF[lo] * (1.f - w) + sFrqF[hi] * w;
  }
  sScan[t] = sum;
  __syncthreads();
  for (int off = 1; off < 1024; off <<= 1) {              // Hillis-Steele inclusive scan
    const float v = (t >= off) ? sScan[t - off] : 0.f;
    __syncthreads();
    sScan[t] += v;
    __syncthreads();
  }
  float run = sScan[t] - sum;                             // exclusive prefix

  for (int i = 0; i < SPT; ++i) {
    const int s = s0 + i;
    float pos = ((float)s + 0.5f) * scale - 0.5f;
    pos = fminf(fmaxf(pos, 0.f), (float)(N_FRAMES - 1));
    const int lo = (int)pos;
    const int hi = min(lo + 1, N_FRAMES - 1);
    const float w = pos - (float)lo;
    run += sFrqF[lo] * (1.f - w) + sFrqF[hi] * w;
    ws_phase[(size_t)b * N_SAMPLES + s] = run - floorf(run);   // frac (freq >= 0)
    ws_envs [(size_t)b * N_SAMPLES + s] = sEnvF[lo] * (1.f - w) + sEnvF[hi] * w;
  }
}

// ---------------------------------------------------------------------------
// Kernel 3: the bandwidth-bound pass. One wave per (b, s): compute the 512
// Gaussian kernel values (v_exp_f32), store them with non-temporal 128-bit
// coalesced writes (512 MB stream, never re-read -> keep it out of the 192 MB
// L2), and fuse the table matvec + env/value scaling for `sampled`.
// ---------------------------------------------------------------------------
__global__ __launch_bounds__(256)
void synth_kernel(const float* __restrict__ ws_table, const float* __restrict__ ws_phase,
                  const float* __restrict__ ws_envs, const float* __restrict__ ws_values,
                  float* __restrict__ out_sampled, float* __restrict__ out_kernel)
{
  __shared__ float sTab[TABLE_SIZE];
  const int b     = blockIdx.x >> 12;          // 4096 blocks per batch item
  const int sBase = (blockIdx.x & 4095) * 8;   // 8 waves -> 8 samples per block
  const int tid   = threadIdx.x;
  const int wave  = tid >> 5, lane = tid & 31;

  for (int j = tid; j < TABLE_SIZE; j += 256)
    sTab[j] = ws_table[b * TABLE_SIZE + j];
  __syncthreads();

  const int    s  = sBase + wave;
  const size_t bs = (size_t)b * N_SAMPLES + s;
  const float  p  = ws_phase[bs];
  const float  INVN = 39.894228040143274f;     // 1 / (STD * sqrt(2*pi)), STD=0.01

  float acc = 0.f;
  float* kout = out_kernel + bs * TABLE_SIZE;
#pragma unroll
  for (int i = 0; i < 4; ++i) {
    const int t0 = i * 128 + lane * 4;         // lanes cover 128 consecutive t -> 512B/wave store
    v4f kv;
#pragma unroll
    for (int j = 0; j < 4; ++j) {
      const int t = t0 + j;
      const float z = ((float)t * (1.f / 511.f) - p) * 100.f;   // (grid - phase)/STD
      const float e = __expf(-0.5f * z * z) * INVN;
      kv[j] = e;
      acc += sTab[t] * e;
    }
    __builtin_nontemporal_store(kv, (v4f*)(kout + t0));  // NT streaming store
  }
  for (int off = 16; off > 0; off >>= 1) acc += __shfl_xor(acc, off, 32);
  if (lane == 0) {
    const float r = acc * ws_envs[bs] * ws_values[b];
    __builtin_nontemporal_store(r, out_sampled + bs);
  }
}

// ---------------------------------------------------------------------------
extern "C" void kernel_launch(void* const* d_in, const int* in_sizes, int n_in,
                              void* d_out, int out_size, void* d_ws, size_t ws_size,
                              hipStream_t stream)
{
  (void)in_sizes; (void)n_in; (void)out_size; (void)ws_size;
  const float* x  = (const float*)d_in[0];
  const float* wt = (const float*)d_in[1];
  const float* p[18];
  for (int i = 0; i < 18; ++i) p[i] = (const float*)d_in[2 + i];

  float* ws        = (float*)d_ws;
  float* ws_values = ws;                          // 8
  float* ws_envf   = ws_values + 8;               // 8*128
  float* ws_frqf   = ws_envf + 8 * N_FRAMES;      // 8*128
  float* ws_table  = ws_frqf + 8 * N_FRAMES;      // 8*512
  float* ws_phase  = ws_table + 8 * TABLE_SIZE;   // 8*32768
  float* ws_envs   = ws_phase + 8 * N_SAMPLES;    // 8*32768  (~2.1 MB total)

  float* out          = (float*)d_out;
  float* out_sampled  = out;                      // (8,1,32768) flat
  float* out_kernel   = out + 8 * N_SAMPLES;      // (8,32768,512) flat

  heads_kernel<<<1, 512, 0, stream>>>(x, wt,
      p[0], p[1], p[2], p[3], p[4], p[5],
      p[6], p[7], p[8], p[9], p[10], p[11],
      p[12], p[13], p[14], p[15], p[16], p[17],
      ws_values, ws_envf, ws_frqf, ws_table);

  scan_kernel<<<8, 1024, 0, stream>>>(ws_envf, ws_frqf, ws_phase, ws_envs);

  synth_kernel<<<8 * 4096, 256, 0, stream>>>(ws_table, ws_phase, ws_envs,
                                             ws_values, out_sampled, out_kernel);
}